// LearnableLogOptimalTransport_24017457119750
// MI455X (gfx1250) — compile-verified
//
#include <hip/hip_runtime.h>
#include <cstdint>
#include <cstddef>

#ifndef __has_builtin
#define __has_builtin(x) 0
#endif

// Problem constants (match reference setup_inputs)
#define BB 4
#define MM 1024
#define NN 1024
#define MP 1025
#define NP 1025
#define SP 1028              // padded per-batch vector stride (16B multiple)
#define NEG_INF (-1000000000000.0f)
#define NCHUNK 8
#define CHUNK_ROWS 129       // ceil(1025/8)

typedef __attribute__((ext_vector_type(2))) float v2f;
typedef __attribute__((ext_vector_type(8))) float v8f;

#if __has_builtin(__builtin_amdgcn_global_load_async_to_lds_b128)
#define HAVE_ASYNC_LDS 1
// exact pointee type per the hipcc diagnostic: int __attribute__((vector_size(16)))
typedef int v4i_b128 __attribute__((vector_size(16)));
typedef __attribute__((address_space(1))) v4i_b128* gptr_b128;
typedef __attribute__((address_space(3))) v4i_b128* lptr_b128;
#else
#define HAVE_ASYNC_LDS 0
#endif

#if __has_builtin(__builtin_amdgcn_s_wait_asynccnt)
#define ASYNC_WAIT() __builtin_amdgcn_s_wait_asynccnt(0)
#else
#define ASYNC_WAIT() asm volatile("s_wait_asynccnt 0" ::: "memory")
#endif

// ---------------------------------------------------------------------------
// K0: per-batch stats: num_valid_row/col, norm, log_mu, log_nu
// ---------------------------------------------------------------------------
__global__ void init_stats_kernel(const unsigned char* __restrict__ rowm,
                                  const unsigned char* __restrict__ colm,
                                  float* __restrict__ log_mu,
                                  float* __restrict__ log_nu,
                                  float* __restrict__ normArr) {
    int b = blockIdx.x;
    int t = threadIdx.x;
    __shared__ float red[256];

    float cnt = 0.f;
    for (int i = t; i < MM; i += 256) cnt += rowm[b * MM + i] ? 1.f : 0.f;
    red[t] = cnt; __syncthreads();
    for (int s = 128; s > 0; s >>= 1) { if (t < s) red[t] += red[t + s]; __syncthreads(); }
    float nr = red[0]; __syncthreads();

    cnt = 0.f;
    for (int j = t; j < NN; j += 256) cnt += colm[b * NN + j] ? 1.f : 0.f;
    red[t] = cnt; __syncthreads();
    for (int s = 128; s > 0; s >>= 1) { if (t < s) red[t] += red[t + s]; __syncthreads(); }
    float nc = red[0]; __syncthreads();

    float norm = -__logf(nr + nc);
    if (t == 0) normArr[b] = norm;

    for (int i = t; i < MP; i += 256) {
        float lm = (i < MM) ? (rowm[b * MM + i] ? norm : NEG_INF)
                            : (__logf(nc) + norm);
        log_mu[b * SP + i] = lm;
    }
    for (int j = t; j < NP; j += 256) {
        float ln = (j < NN) ? (colm[b * NN + j] ? norm : NEG_INF)
                            : (__logf(nr) + norm);
        log_nu[b * SP + j] = ln;
    }
}

// ---------------------------------------------------------------------------
// K1: materialize padded+masked score matrix P (into d_out; L2-resident after)
// ---------------------------------------------------------------------------
__global__ void build_p_kernel(const float* __restrict__ scores,
                               const unsigned char* __restrict__ rowm,
                               const unsigned char* __restrict__ colm,
                               const float* __restrict__ alphaPtr,
                               float* __restrict__ P) {
    int i = blockIdx.x, b = blockIdx.y, t = threadIdx.x;
    float alpha = alphaPtr[0];
    bool rvalid = (i < MM) ? (rowm[b * MM + i] != 0) : true;
    size_t rowoff = ((size_t)b * MP + i) * NP;
    for (int j = t; j < NP; j += 256) {
        bool cvalid = (j < NN) ? (colm[b * NN + j] != 0) : true;
        float val = (i < MM && j < NN) ? scores[((size_t)b * MM + i) * NN + j] : alpha;
        P[rowoff + j] = (rvalid && cvalid) ? val : NEG_INF;
    }
}

// ---------------------------------------------------------------------------
// K2 (u pass): one wave per padded row. Builds v in LDS from per-chunk
// partials (iter 0: v=0), then u = log_mu - log(sum_j exp(P + v)).
// Main loop is a fixed 32-trip branchless stream (1024 = 32 lanes x 32).
// ---------------------------------------------------------------------------
__global__ void u_pass_kernel(const float* __restrict__ P,
                              const float* __restrict__ colPart,
                              const float* __restrict__ log_mu,
                              const float* __restrict__ log_nu,
                              float* __restrict__ u,
                              int iter) {
    __shared__ __attribute__((aligned(16))) float sv[SP];
    int b = blockIdx.y, t = threadIdx.x;

    for (int j = t; j < NP; j += 256) {
        float vj = 0.f;
        if (iter != 0) {
            float s = 0.f;
            #pragma unroll
            for (int k = 0; k < NCHUNK; ++k)
                s += colPart[(size_t)(k * BB + b) * SP + j];
            vj = log_nu[b * SP + j] - __logf(fmaxf(s, 1e-30f));
        }
        sv[j] = vj;
    }
    __syncthreads();

    int wave = t >> 5, lane = t & 31;
    int i = blockIdx.x * 8 + wave;
    if (i < MP) {
        const float* row = P + ((size_t)b * MP + i) * NP;
        float sum = 0.f;
        #pragma unroll 4
        for (int j = lane; j < 1024; j += 32)       // exactly 32 trips, no divergence
            sum += __expf(row[j] + sv[j]);          // masked (-1e12) underflows to 0
        if (lane == 0) sum += __expf(row[1024] + sv[1024]);
        for (int off = 16; off > 0; off >>= 1) sum += __shfl_xor(sum, off, 32);
        if (lane == 0)
            u[b * SP + i] = log_mu[b * SP + i] - __logf(fmaxf(sum, 1e-30f));
    }
}

// ---------------------------------------------------------------------------
// K3 (v pass, WMMA): wave handles 16 columns x one row chunk.
//   B(4x16 f32) = exp(P + u) tile, A = ones(16x4) => C rows = column sums,
//   exact f32 on the matrix pipe while the VALU streams v_exp_f32.
//   Peeled main loop (8 rows, 2 independent accumulators) is branchless;
//   tail uses clamped indices + v_cndmask zeroing (no EXEC divergence).
//   u vector is staged to LDS via GLOBAL_LOAD_ASYNC_TO_LDS_B128 when available.
// ---------------------------------------------------------------------------
__global__ void v_pass_kernel(const float* __restrict__ P,
                              const float* __restrict__ u,
                              float* __restrict__ colPart) {
    __shared__ __attribute__((aligned(16))) float su[SP];
    int b = blockIdx.z, t = threadIdx.x;

#if HAVE_ASYNC_LDS
    {
        // 256 threads x one b128 = floats 0..1023 (u + b*SP + 4t is 16B aligned)
        float* gsrc = const_cast<float*>(u) + (size_t)b * SP + 4 * t;
        __builtin_amdgcn_global_load_async_to_lds_b128(
            (gptr_b128)gsrc, (lptr_b128)&su[4 * t], 0, 0);
        if (t == 0) su[1024] = u[(size_t)b * SP + 1024];
        ASYNC_WAIT();
    }
#else
    for (int i = t; i < MP; i += 256) su[i] = u[b * SP + i];
#endif
    __syncthreads();

    int wave = t >> 5, lane = t & 31;
    int j0 = (blockIdx.x * 8 + wave) * 16;
    if (j0 >= NP) return;                      // wave-uniform: EXEC stays full

    int chunk = blockIdx.y;
    int r0 = chunk * CHUNK_ROWS;
    int rEnd = r0 + CHUNK_ROWS; if (rEnd > MP) rEnd = MP;
    int nrows = rEnd - r0;

    int col = j0 + (lane & 15);
    int cc = (col < NP) ? col : (NP - 1);      // clamp: garbage cols never written
    int half = (lane >> 4) & 1;

    const float* Pb = P + (size_t)b * MP * NP;
    v2f a = {1.f, 1.f};
    v8f c0 = {0.f, 0.f, 0.f, 0.f, 0.f, 0.f, 0.f, 0.f};
    v8f c1 = c0;

    int r = r0;
    int end8 = r0 + (nrows & ~7);
    for (; r < end8; r += 8) {                 // branchless: 8 rows, 2 indep WMMAs
        int pr = (r + 16 < MP) ? r + 16 : r;
        __builtin_prefetch(Pb + (size_t)pr * NP + cc, 0, 3);
        int ra = r + half;
        float e0 = __expf(Pb[(size_t)ra * NP + cc] + su[ra]);
        float e1 = __expf(Pb[(size_t)(ra + 2) * NP + cc] + su[ra + 2]);
        v2f bm0 = {e0, e1};
        c0 = __builtin_amdgcn_wmma_f32_16x16x4_f32(false, a, false, bm0,
                                                   (short)0, c0, false, false);
        float e2 = __expf(Pb[(size_t)(ra + 4) * NP + cc] + su[ra + 4]);
        float e3 = __expf(Pb[(size_t)(ra + 6) * NP + cc] + su[ra + 6]);
        v2f bm1 = {e2, e3};
        c1 = __builtin_amdgcn_wmma_f32_16x16x4_f32(false, a, false, bm1,
                                                   (short)0, c1, false, false);
    }
    for (; r < rEnd; r += 4) {                 // tail: clamped loads, select-zeroed
        int ra = r + half;
        int rb = ra + 2;
        int rca = (ra < rEnd) ? ra : r0;
        int rcb = (rb < rEnd) ? rb : r0;
        float e0 = __expf(Pb[(size_t)rca * NP + cc] + su[rca]);
        float e1 = __expf(Pb[(size_t)rcb * NP + cc] + su[rcb]);
        e0 = (ra < rEnd) ? e0 : 0.f;
        e1 = (rb < rEnd) ? e1 : 0.f;
        v2f bm = {e0, e1};
        c0 = __builtin_amdgcn_wmma_f32_16x16x4_f32(false, a, false, bm,
                                                   (short)0, c0, false, false);
    }

    // C vgpr0: lane L<16 holds (M=0, N=L) = colsum for column j0+L
    float s = c0[0] + c1[0];
    if (lane < 16 && col < NP)
        colPart[(size_t)(chunk * BB + b) * SP + col] = s;
}

// ---------------------------------------------------------------------------
// K4: finalize v from last iteration's partials
// ---------------------------------------------------------------------------
__global__ void finalize_v_kernel(const float* __restrict__ colPart,
                                  const float* __restrict__ log_nu,
                                  float* __restrict__ v) {
    int idx = blockIdx.x * 256 + threadIdx.x;
    if (idx < BB * NP) {
        int b = idx / NP, j = idx % NP;
        float s = 0.f;
        #pragma unroll
        for (int k = 0; k < NCHUNK; ++k)
            s += colPart[(size_t)(k * BB + b) * SP + j];
        v[b * SP + j] = log_nu[b * SP + j] - __logf(fmaxf(s, 1e-30f));
    }
}

// ---------------------------------------------------------------------------
// K5: out = P + u + v - norm (in place on d_out)
// ---------------------------------------------------------------------------
__global__ void output_kernel(float* __restrict__ P,
                              const float* __restrict__ u,
                              const float* __restrict__ v,
                              const float* __restrict__ normArr) {
    int i = blockIdx.x, b = blockIdx.y, t = threadIdx.x;
    float add_i = u[b * SP + i] - normArr[b];
    size_t rowoff = ((size_t)b * MP + i) * NP;
    for (int j = t; j < NP; j += 256)
        P[rowoff + j] = P[rowoff + j] + add_i + v[b * SP + j];
}

// ---------------------------------------------------------------------------
extern "C" void kernel_launch(void* const* d_in, const int* in_sizes, int n_in,
                              void* d_out, int out_size, void* d_ws, size_t ws_size,
                              hipStream_t stream) {
    const float* scores        = (const float*)d_in[0];
    const unsigned char* rowm  = (const unsigned char*)d_in[1];
    const unsigned char* colm  = (const unsigned char*)d_in[2];
    const float* alpha         = (const float*)d_in[3];

    float* P  = (float*)d_out;          // padded scores live in d_out (L2-resident)
    float* ws = (float*)d_ws;
    float* log_mu  = ws;                          // BB*SP
    float* log_nu  = log_mu + BB * SP;            // BB*SP
    float* u       = log_nu + BB * SP;            // BB*SP (16B-aligned slices)
    float* v       = u + BB * SP;                 // BB*SP
    float* normArr = v + BB * SP;                 // 8 floats (keeps 16B alignment)
    float* colPart = normArr + 8;                 // NCHUNK*BB*SP

    init_stats_kernel<<<BB, 256, 0, stream>>>(rowm, colm, log_mu, log_nu, normArr);
    build_p_kernel<<<dim3(MP, BB), 256, 0, stream>>>(scores, rowm, colm, alpha, P);

    for (int it = 0; it < 100; ++it) {
        u_pass_kernel<<<dim3(129, BB), 256, 0, stream>>>(P, colPart, log_mu, log_nu, u, it);
        v_pass_kernel<<<dim3(9, NCHUNK, BB), 256, 0, stream>>>(P, u, colPart);
    }

    finalize_v_kernel<<<(BB * NP + 255) / 256, 256, 0, stream>>>(colPart, log_nu, v);
    output_kernel<<<dim3(MP, BB), 256, 0, stream>>>(P, u, v, normArr);
}